// Perspective3d_70686571758254
// MI455X (gfx1250) — compile-verified
//
#include <hip/hip_runtime.h>
#include <hip/hip_bf16.h>

// ---------------- problem constants (fixed by setup_inputs) ----------------
#define NB   4      // batch
#define NC   32     // channels
#define VD   64     // voxel D == H == W
#define XY   96     // xy_sample
#define ZS   48     // z_sample
#define FXY  100.0f
#define UXY  48.0f

constexpr int   HW        = XY * XY;            // 9216
constexpr int   PTS_PER_B = HW * ZS;            // 442368 (divisible by 256)
constexpr int   TOTAL_PTS = NB * PTS_PER_B;     // 1769472
constexpr int   CH_STRIDE = VD * VD * VD;       // 262144 (voxel channel stride)
constexpr float STEP_XY   = 96.0f / 95.0f;      // linspace(0,96,96) step
constexpr float STEP_Z    = 1.0f / 47.0f;       // linspace(-0.5,0.5,48) step

typedef __attribute__((ext_vector_type(2))) float v2f;
typedef __attribute__((ext_vector_type(8))) float v8f;

// ---------------------------------------------------------------------------
// Setup kernel: one wave32. Builds rt_inv[b] and scale_inv[b] diagonals in
// LDS, then computes all four wTc[b] = scale_inv[b] @ rt_inv[b] with a single
// V_WMMA_F32_16X16X4_F32 (batches packed block-diagonally: A[4b+i,k] =
// scale_inv[b][i][k], B[k,4b+j] = rt_inv[b][k][j]; diagonal 4x4 blocks of D
// are the per-batch products). Result -> ws as wtc[b][i][j] row-major.
// ---------------------------------------------------------------------------
__global__ __launch_bounds__(32)
void camera2world_wmma_kernel(const float* __restrict__ rot,   // [B,4,4]
                              const float* __restrict__ tran,  // [B,3]
                              const float* __restrict__ s,     // [B,3]
                              float* __restrict__ wtc)         // [B,16] out
{
    __shared__ float rt[NB * 16];   // rt_inv[b][row][col]
    __shared__ float si[16];        // scale_inv diagonal, packed per (b,i)
    __shared__ float dmat[16 * 16]; // full D matrix readback

    const int lane = threadIdx.x;   // 0..31

    // rt_inv: top = [R^T | -R^T t], bottom = [0 0 0 1]
    for (int e = lane; e < NB * 16; e += 32) {
        const int b  = e >> 4;
        const int rr = (e >> 2) & 3;   // row
        const int cc = e & 3;          // col
        float v;
        if (rr == 3) {
            v = (cc == 3) ? 1.0f : 0.0f;
        } else if (cc < 3) {
            v = rot[b * 16 + cc * 4 + rr];          // R^T[rr][cc] = R[cc][rr]
        } else {
            v = -(rot[b * 16 + 0 + rr] * tran[b * 3 + 0] +
                  rot[b * 16 + 4 + rr] * tran[b * 3 + 1] +
                  rot[b * 16 + 8 + rr] * tran[b * 3 + 2]);
        }
        rt[e] = v;
    }
    if (lane < 16) {
        const int b = lane >> 2, i = lane & 3;
        si[lane] = (i < 3) ? 1.0f / s[b * 3 + i] : 1.0f;
    }
    __syncthreads();

    // A fragment (16x4 f32, MxK): lanes 0-15 -> K=0,1 ; lanes 16-31 -> K=2,3
    const int m  = lane & 15;           // A row = 4b + i
    const int iA = m & 3;
    const int k0 = (lane < 16) ? 0 : 2;
    v2f a;
    a.x = (iA == k0    ) ? si[m] : 0.0f;   // A[m][k0]
    a.y = (iA == k0 + 1) ? si[m] : 0.0f;   // A[m][k0+1]

    // B fragment (4x16 f32, KxN): same half-wave K split as A (assumed)
    const int n  = lane & 15;           // B col = 4b + j
    const int bB = n >> 2, jB = n & 3;
    v2f bm;
    bm.x = rt[bB * 16 + (k0    ) * 4 + jB];  // B[k0][n]
    bm.y = rt[bB * 16 + (k0 + 1) * 4 + jB];  // B[k0+1][n]

    v8f cz = {};
    // D = A x B + 0  ->  v_wmma_f32_16x16x4_f32
    v8f d = __builtin_amdgcn_wmma_f32_16x16x4_f32(
        /*neg_a=*/false, a, /*neg_b=*/false, bm,
        /*c_mod=*/(short)0, cz, /*reuse_a=*/false, /*reuse_b=*/false);

    // D layout: VGPR v -> M = v (lanes 0-15) / v+8 (lanes 16-31), N = lane&15
#pragma unroll
    for (int v = 0; v < 8; ++v)
        dmat[(v + 8 * (lane >> 4)) * 16 + (lane & 15)] = d[v];
    __syncthreads();

    // extract diagonal 4x4 blocks -> wtc[b][i][j]
    for (int e = lane; e < NB * 16; e += 32) {
        const int b = e >> 4, i = (e >> 2) & 3, j = e & 3;
        wtc[e] = dmat[(b * 4 + i) * 16 + (b * 4 + j)];
    }
}

// ---------------------------------------------------------------------------
// Main kernel: one thread per spatial point (b,z,y,x); compute camera ray,
// transform by wTc, trilinear weights/offsets once, then loop the 32 channels.
// Output streamed with non-temporal stores to preserve L2 for the voxel set.
// ---------------------------------------------------------------------------
__global__ __launch_bounds__(256)
void perspective_sample_kernel(const float* __restrict__ vox,   // [B,C,64,64,64]
                               const float* __restrict__ tran,  // [B,3]
                               const float* __restrict__ det,   // [B,2]
                               const float* __restrict__ wtc,   // [B,16]
                               float* __restrict__ out)         // [B,C,48,96,96]
{
    const int idx = blockIdx.x * 256 + threadIdx.x;
    if (idx >= TOTAL_PTS) return;

    // batch index is uniform per block (PTS_PER_B % 256 == 0) -> scalar loads
    const int b = (int)((blockIdx.x * 256u) / (unsigned)PTS_PER_B);
    const int r = idx - b * PTS_PER_B;
    const int x  = r % XY;
    const int t2 = r / XY;
    const int y  = t2 % XY;
    const int z  = t2 / XY;

    const float tz = tran[b * 3 + 2];
    const float dx = det[b * 2 + 0];
    const float dy = det[b * 2 + 1];
    float mmat[16];
#pragma unroll
    for (int i = 0; i < 16; ++i) mmat[i] = wtc[b * 16 + i];

    // camera-space ray point
    const float xt = x * STEP_XY;
    const float yt = y * STEP_XY;
    const float zt = -0.5f + z * STEP_Z;
    const float Zt = zt + tz;
    const float Xt = (xt - UXY + dx) * Zt * (1.0f / FXY);
    const float Yt = (yt - UXY + dy) * Zt * (1.0f / FXY);

    // world = wTc * cam  (wGrid[j] = sum_k wTc[j][k] * cam[k])
    const float w0 = mmat[0] * Xt + mmat[1] * Yt + mmat[2]  * Zt + mmat[3];
    const float w1 = mmat[4] * Xt + mmat[5] * Yt + mmat[6]  * Zt + mmat[7];
    const float w2 = mmat[8] * Xt + mmat[9] * Yt + mmat[10] * Zt + mmat[11];
    const float w3 = mmat[12]* Xt + mmat[13]* Yt + mmat[14] * Zt + mmat[15];
    const float inv = 2.0f / w3;
    const float gx = w0 * inv, gy = w1 * inv, gz = w2 * inv;

    // grid_sample, align_corners=True, zeros padding; all dims = 64
    const float ix = (gx + 1.0f) * 31.5f;
    const float iy = (gy + 1.0f) * 31.5f;
    const float iz = (gz + 1.0f) * 31.5f;
    const float fx0 = floorf(ix), fy0 = floorf(iy), fz0 = floorf(iz);
    const float fx = ix - fx0, fy = iy - fy0, fz = iz - fz0;
    const int ix0 = (int)fx0, iy0 = (int)fy0, iz0 = (int)fz0;

    float wgt[8];
    int   off[8];
#pragma unroll
    for (int dzc = 0; dzc < 2; ++dzc)
#pragma unroll
        for (int dyc = 0; dyc < 2; ++dyc)
#pragma unroll
            for (int dxc = 0; dxc < 2; ++dxc) {
                const int k  = dzc * 4 + dyc * 2 + dxc;
                const int xi = ix0 + dxc, yi = iy0 + dyc, zi = iz0 + dzc;
                const bool valid = (xi >= 0) & (xi <= VD - 1) &
                                   (yi >= 0) & (yi <= VD - 1) &
                                   (zi >= 0) & (zi <= VD - 1);
                const int xc = min(max(xi, 0), VD - 1);
                const int yc = min(max(yi, 0), VD - 1);
                const int zc = min(max(zi, 0), VD - 1);
                const float wv = (dzc ? fz : 1.0f - fz) *
                                 (dyc ? fy : 1.0f - fy) *
                                 (dxc ? fx : 1.0f - fx);
                wgt[k] = valid ? wv : 0.0f;
                off[k] = (zc * VD + yc) * VD + xc;
            }

    const float* vb    = vox + (size_t)b * NC * CH_STRIDE;
    const size_t obase = ((size_t)(b * NC) * ZS + z) * HW + y * XY + x;

#pragma unroll 8
    for (int c = 0; c < NC; ++c) {
        const float* vc = vb + (size_t)c * CH_STRIDE;
        float acc = wgt[0] * vc[off[0]] + wgt[1] * vc[off[1]] +
                    wgt[2] * vc[off[2]] + wgt[3] * vc[off[3]] +
                    wgt[4] * vc[off[4]] + wgt[5] * vc[off[5]] +
                    wgt[6] * vc[off[6]] + wgt[7] * vc[off[7]];
        __builtin_nontemporal_store(acc, out + obase + (size_t)c * (ZS * HW));
    }
}

// ---------------------------------------------------------------------------
extern "C" void kernel_launch(void* const* d_in, const int* in_sizes, int n_in,
                              void* d_out, int out_size, void* d_ws, size_t ws_size,
                              hipStream_t stream)
{
    const float* voxels = (const float*)d_in[0];  // [4,32,64,64,64]
    const float* rot    = (const float*)d_in[1];  // [4,4,4]
    const float* tran   = (const float*)d_in[2];  // [4,3]
    const float* s      = (const float*)d_in[3];  // [4,3]
    const float* det    = (const float*)d_in[4];  // [4,2]
    // d_in[5]=xy_sample(96), d_in[6]=z_sample(48): fixed, hardcoded above
    float* out = (float*)d_out;
    float* wtc = (float*)d_ws;                    // 64 floats of scratch

    camera2world_wmma_kernel<<<1, 32, 0, stream>>>(rot, tran, s, wtc);

    const int nblk = TOTAL_PTS / 256;             // 6912, exact
    perspective_sample_kernel<<<nblk, 256, 0, stream>>>(voxels, tran, det, wtc, out);
}